// MarginLoss_59287728554367
// MI455X (gfx1250) — compile-verified
//
#include <hip/hip_runtime.h>
#include <stdint.h>

#define MARG   0.5f
#define NW     8      // waves per block
#define TCOLS  32     // columns per LDS tile
#define WCOLS  64     // columns per wave (2 tiles)
#define SRD    36     // LDS row stride in dwords (36 = 0 mod 4 -> b128 aligned; 4r+c bank map conflict-free)
#define CSPLIT 8      // column splits across blocks

static const int Bsz = 4096, Dsz = 4096, Psz = 8;

typedef float v2f __attribute__((ext_vector_type(2)));
typedef float v8f __attribute__((ext_vector_type(8)));

// Issue one 16-row x 32-dword tile copy (global -> LDS) as 4 async b128 instructions.
__device__ __forceinline__ void async_tile(const float* gbase, unsigned lds_base) {
  const int lane = threadIdx.x & 31;
#pragma unroll
  for (int j = 0; j < 4; ++j) {
    int unit  = j * 32 + lane;   // 128 transfers of 16B: 16 rows x 8 chunks
    int row   = unit >> 3;
    int chunk = unit & 7;
    unsigned lds  = lds_base + (unsigned)(row * (SRD * 4) + chunk * 16);
    unsigned goff = (unsigned)(row * Dsz * 4 + chunk * 16);
    asm volatile("global_load_async_to_lds_b128 %0, %1, %2"
                 :: "v"(lds), "v"(goff), "s"(gbase)
                 : "memory");
  }
}

__device__ __forceinline__ void compute_tile(const float* tb, int row,
                                             float ck0, float ck1, float ck2, float ck3,
                                             v8f& acc1, v8f& acc2, v2f ones) {
#pragma unroll
  for (int c = 0; c < TCOLS; c += 4) {
    float4 xv = *reinterpret_cast<const float4*>(tb + row * SRD + c);
    float xd[4] = {xv.x, xv.y, xv.z, xv.w};
#pragma unroll
    for (int j = 0; j < 4; ++j) {
      v2f a, b;
      a.x = fmaxf(xd[j] + ck0, 0.f);
      a.y = fmaxf(xd[j] + ck1, 0.f);
      b.x = fmaxf(xd[j] + ck2, 0.f);
      b.y = fmaxf(xd[j] + ck3, 0.f);
      acc1 = __builtin_amdgcn_wmma_f32_16x16x4_f32(false, a, false, ones, (short)0, acc1, false, false);
      acc2 = __builtin_amdgcn_wmma_f32_16x16x4_f32(false, b, false, ones, (short)0, acc2, false, false);
    }
  }
}

__global__ __launch_bounds__(256)
void margin_main(const float* __restrict__ x, const int* __restrict__ pos,
                 float* __restrict__ partials) {
  __shared__ float xs[NW * 2 * 16 * SRD];   // per-wave double buffers, 36 KB
  __shared__ float cvals[16][Psz];
  __shared__ int   pids[16][Psz];
  __shared__ float corr_arr[16];
  __shared__ float wsum[NW];

  const int tid  = threadIdx.x;
  const int lane = tid & 31;
  const int wave = tid >> 5;
  const int r0   = blockIdx.x * 16;
  const int c0w  = blockIdx.y * (NW * WCOLS) + wave * WCOLS;

  // Gather positive scores, build per-row constants c_k = margin - x_pos.
  if (tid < 16 * Psz) {
    int r = tid >> 3, k = tid & 7;
    int p = pos[(r0 + r) * Psz + k];
    float xp = x[(size_t)(r0 + r) * Dsz + p];
    cvals[r][k] = MARG - xp;
    pids[r][k]  = p;
  }
  __syncthreads();

  // Mask correction (only once per row group): subtract positive-column violations.
  if (blockIdx.y == 0 && tid < 16) {
    int r = tid;
    float corr = 0.f;
    for (int j = 0; j < Psz; ++j) {
      bool first = true;
      for (int j2 = 0; j2 < j; ++j2) first = first && (pids[r][j2] != pids[r][j]);
      if (first) {
        float xj = MARG - cvals[r][j];       // = x_pos[r][j]
        for (int k = 0; k < Psz; ++k) corr += fmaxf(xj + cvals[r][k], 0.f);
      }
    }
    corr_arr[r] = corr;
  }

  // Kick off async staging of both tiles for this wave.
  const float* gx = x + (size_t)r0 * Dsz + c0w;
  unsigned lds0 = (unsigned)(uintptr_t)(xs + wave * 2 * 16 * SRD);
  unsigned lds1 = lds0 + 16 * SRD * 4;
  async_tile(gx, lds0);
  async_tile(gx + TCOLS, lds1);

  const int row  = lane & 15;
  const int half = lane >> 4;
  float ck0 = cvals[row][half * 2 + 0];
  float ck1 = cvals[row][half * 2 + 1];
  float ck2 = cvals[row][4 + half * 2 + 0];
  float ck3 = cvals[row][4 + half * 2 + 1];

  v8f acc1 = {}; v8f acc2 = {};
  v2f ones; ones.x = 1.f; ones.y = 1.f;

  const float* xsw = xs + wave * 2 * 16 * SRD;

  asm volatile("s_wait_asynccnt 0x4" ::: "memory");   // tile 0 landed
  compute_tile(xsw, row, ck0, ck1, ck2, ck3, acc1, acc2, ones);
  asm volatile("s_wait_asynccnt 0x0" ::: "memory");   // tile 1 landed
  compute_tile(xsw + 16 * SRD, row, ck0, ck1, ck2, ck3, acc1, acc2, ones);

  // C columns replicate row sums -> sum everything, divide by 16 later.
  float s = 0.f;
#pragma unroll
  for (int j = 0; j < 8; ++j) s += acc1[j] + acc2[j];
#pragma unroll
  for (int off = 16; off > 0; off >>= 1) s += __shfl_down(s, off, 32);
  if (lane == 0) wsum[wave] = s;
  __syncthreads();

  if (tid == 0) {
    float t = 0.f;
    for (int w = 0; w < NW; ++w) t += wsum[w];
    t *= (1.f / 16.f);
    if (blockIdx.y == 0) {
      float c = 0.f;
      for (int r = 0; r < 16; ++r) c += corr_arr[r];
      t -= c;
    }
    partials[blockIdx.x * gridDim.y + blockIdx.y] = t;
  }
}

__global__ __launch_bounds__(256)
void reduce_final(const float* __restrict__ partials, int n, float* __restrict__ out) {
  __shared__ float ws[8];
  float s = 0.f;
  for (int i = threadIdx.x; i < n; i += 256) s += partials[i];
#pragma unroll
  for (int off = 16; off > 0; off >>= 1) s += __shfl_down(s, off, 32);
  if ((threadIdx.x & 31) == 0) ws[threadIdx.x >> 5] = s;
  __syncthreads();
  if (threadIdx.x == 0) {
    float t = 0.f;
    for (int w = 0; w < 8; ++w) t += ws[w];
    double count = (double)Bsz * (double)Psz * (double)(Dsz - Psz);
    out[0] = (float)((double)t / count);
  }
}

extern "C" void kernel_launch(void* const* d_in, const int* in_sizes, int n_in,
                              void* d_out, int out_size, void* d_ws, size_t ws_size,
                              hipStream_t stream) {
  const float* x   = (const float*)d_in[0];
  // d_in[1] = y (multi-hot labels): redundant given pos_ids, intentionally unread.
  const int*   pos = (const int*)d_in[2];
  float* partials  = (float*)d_ws;
  (void)in_sizes; (void)n_in; (void)out_size; (void)ws_size;

  dim3 grid(Bsz / 16, CSPLIT);
  margin_main<<<grid, 256, 0, stream>>>(x, pos, partials);
  reduce_final<<<1, 256, 0, stream>>>(partials, (Bsz / 16) * CSPLIT, (float*)d_out);
}